// AttentionPairBias_4526895530229
// MI455X (gfx1250) — compile-verified
//
#include <hip/hip_runtime.h>
#include <hip/hip_bf16.h>
#include <math.h>

// ---------------------------------------------------------------------------
// AttentionPairBias for MI455X (gfx1250), fp32 end-to-end using
// V_WMMA_F32_16X16X4_F32 (exact fp32 accumulate, wave32).
//
// Sizes (compile-time): B=1, N=768, DT=768, DP=128, DS=384, H=16, D=48.
//
// WMMA f32 16x16x4 fragment layout assumptions (ISA 7.12.2):
//   A (16x4):  v2f per lane; lane L<16 -> A[L][0..1], lane L>=16 -> A[L-16][2..3]
//   B (4x16):  v2f per lane; element v of lane L -> B[v + 2*(L/16)][L%16]
//   C (16x16): v8f per lane; element r of lane L -> C[r + 8*(L/16)][L%16]
// ---------------------------------------------------------------------------

typedef __attribute__((ext_vector_type(2))) float v2f;
typedef __attribute__((ext_vector_type(8))) float v8f;

#define NTOK 768
#define DTOT 768
#define DSNG 384
#define DPAIR 128
#define NHEAD 16
#define DHEAD 48

__device__ __forceinline__ v8f wmma4(v2f a, v2f b, v8f c) {
  // 8 args: (neg_a, A, neg_b, B, c_mod, C, reuse_a, reuse_b)
  return __builtin_amdgcn_wmma_f32_16x16x4_f32(false, a, false, b, (short)0, c,
                                               false, false);
}

__device__ __forceinline__ float sigmoidf_(float x) {
  return 1.0f / (1.0f + expf(-x));
}

// ---------------------------------------------------------------------------
// Row LayerNorm: one block (256 threads) per row of length L.
// y = (x - mean) * rsqrt(var + eps) [* w + b]
// ---------------------------------------------------------------------------
__global__ __launch_bounds__(256) void ln_kernel(const float* __restrict__ x,
                                                 const float* __restrict__ w,
                                                 const float* __restrict__ b,
                                                 float* __restrict__ y, int L) {
  __shared__ float red[256];
  const int tid = threadIdx.x;
  const float* xr = x + (size_t)blockIdx.x * L;
  float* yr = y + (size_t)blockIdx.x * L;

  float s = 0.f;
  for (int i = tid; i < L; i += 256) s += xr[i];
  red[tid] = s;
  __syncthreads();
  for (int off = 128; off > 0; off >>= 1) {
    if (tid < off) red[tid] += red[tid + off];
    __syncthreads();
  }
  const float mean = red[0] / (float)L;
  __syncthreads();

  float s2 = 0.f;
  for (int i = tid; i < L; i += 256) {
    float d = xr[i] - mean;
    s2 += d * d;
  }
  red[tid] = s2;
  __syncthreads();
  for (int off = 128; off > 0; off >>= 1) {
    if (tid < off) red[tid] += red[tid + off];
    __syncthreads();
  }
  const float inv = rsqrtf(red[0] / (float)L + 1e-5f);

  if (w != nullptr) {
    for (int i = tid; i < L; i += 256)
      yr[i] = (xr[i] - mean) * inv * w[i] + b[i];
  } else {
    for (int i = tid; i < L; i += 256) yr[i] = (xr[i] - mean) * inv;
  }
}

// ---------------------------------------------------------------------------
// Generic fp32 WMMA GEMM: C[M,Nc] = epilogue(A[M,K] @ W[K,Nc] + bias)
// Wave computes a 16x64 strip (4 C tiles); block = 8 waves -> 128(M) x 64(N).
// Grid: (M/128, Nc/64). M, Nc multiples of 128/64; K multiple of 4.
// modes: 0: alpha*(acc+bias)
//        1: sigmoid(acc+bias)
//        2: (acc+bias) * sigmoid(e0[idx])
//        3: sigmoid(acc+bias) * e0[idx] + e1[idx]
// ---------------------------------------------------------------------------
__global__ __launch_bounds__(256) void gemm_kernel(
    const float* __restrict__ A, const float* __restrict__ W,
    const float* __restrict__ bias, float* __restrict__ C, int K, int Nc,
    float alpha, int mode, const float* __restrict__ e0,
    const float* __restrict__ e1) {
  const int lane = threadIdx.x & 31;
  const int wave = threadIdx.x >> 5;
  const int lm = lane & 15;
  const int hi = lane >> 4;        // 0 or 1
  const int lk = hi << 1;          // 0 or 2 (K sub-offset of this half-wave)
  const int m0 = blockIdx.x * 128 + wave * 16;
  const int n0 = blockIdx.y * 64;

  v8f acc[4] = {};
  const float* arow = A + (size_t)(m0 + lm) * K + lk;

  for (int k = 0; k < K; k += 4) {
    v2f af = *(const v2f*)(arow + k);
#pragma unroll
    for (int t = 0; t < 4; ++t) {
      const float* wp = W + (size_t)(k + lk) * Nc + (n0 + t * 16 + lm);
      v2f bf;
      bf[0] = wp[0];
      bf[1] = wp[Nc];
      acc[t] = wmma4(af, bf, acc[t]);
    }
  }

#pragma unroll
  for (int t = 0; t < 4; ++t) {
    const int col = n0 + t * 16 + lm;
    const float bv = (bias != nullptr) ? bias[col] : 0.0f;
#pragma unroll
    for (int r = 0; r < 8; ++r) {
      const int row = m0 + r + (hi << 3);
      const size_t idx = (size_t)row * Nc + col;
      const float v = acc[t][r] + bv;
      float outv;
      if (mode == 0)
        outv = alpha * v;
      else if (mode == 1)
        outv = sigmoidf_(v);
      else if (mode == 2)
        outv = v * sigmoidf_(e0[idx]);
      else
        outv = sigmoidf_(v) * e0[idx] + e1[idx];
      C[idx] = outv;
    }
  }
}

// ---------------------------------------------------------------------------
// Pair bias: pb[p*16 + h] = z[p,:] . wb[:,h] + wb_b[h] + beta[p*16+h]
// p = i*N + j flat pair index. Wave handles 16 consecutive pairs (16x16 tile
// over heads), K=128. Block = 8 waves -> 128 pairs; grid = N*N/128 = 4608.
// Streams z (302 MB) exactly once.
// ---------------------------------------------------------------------------
__global__ __launch_bounds__(256) void pair_bias_kernel(
    const float* __restrict__ z, const float* __restrict__ wb,
    const float* __restrict__ wb_b, const float* __restrict__ beta,
    float* __restrict__ pb) {
  const int lane = threadIdx.x & 31;
  const int wave = threadIdx.x >> 5;
  const int lm = lane & 15;
  const int hi = lane >> 4;
  const int lk = hi << 1;
  const size_t p0 = ((size_t)blockIdx.x * 8 + wave) * 16;

  v8f acc = {};
  const float* zrow = z + (p0 + lm) * DPAIR + lk;
#pragma unroll 4
  for (int k = 0; k < DPAIR; k += 4) {
    v2f af = *(const v2f*)(zrow + k);
    const float* wp = wb + (size_t)(k + lk) * NHEAD + lm;
    v2f bf;
    bf[0] = wp[0];
    bf[1] = wp[NHEAD];
    acc = wmma4(af, bf, acc);
  }

  const float bb = wb_b[lm];
#pragma unroll
  for (int r = 0; r < 8; ++r) {
    const size_t p = p0 + r + (hi << 3);
    const size_t idx = p * NHEAD + lm;
    pb[idx] = acc[r] + bb + beta[idx];
  }
}

// ---------------------------------------------------------------------------
// Attention: one block (256 thr, 8 waves) per (query tile of 16, head).
//   logits  P[16][768] = Q(16x48) @ K^T  (WMMA, j-tiles split over waves)
//   softmax over j with pair bias fused (unnormalized exp kept in LDS)
//   O(16x48) = P @ V  (WMMA, K=768 split over waves, LDS float atomics)
//   epilogue: o = g * O / rowsum
// LDS row stride 772 floats -> A-frag ds_read_b64 is bank-conflict free.
// ---------------------------------------------------------------------------
#define PSTR 772
__global__ __launch_bounds__(256) void attn_kernel(
    const float* __restrict__ q, const float* __restrict__ kmat,
    const float* __restrict__ vmat, const float* __restrict__ g,
    const float* __restrict__ pb, float* __restrict__ o) {
  __shared__ float P[16][PSTR];
  __shared__ float accbuf[16][DHEAD];
  __shared__ float red[256];
  __shared__ float rowsum[16];
  __shared__ float rowmax[16];

  const int tid = threadIdx.x;
  const int lane = tid & 31;
  const int wave = tid >> 5;
  const int lm = lane & 15;
  const int hi = lane >> 4;
  const int lk = hi << 1;
  const int i0 = blockIdx.x * 16;   // query tile base
  const int h = blockIdx.y;         // head
  const int ho = h * DHEAD;

  for (int idx = tid; idx < 16 * DHEAD; idx += 256)
    ((float*)accbuf)[idx] = 0.0f;

  // ---- logits: Q(16x48) @ K^T, q already scaled by D^-0.5
  v2f aq[12];
#pragma unroll
  for (int kk = 0; kk < 12; ++kk)
    aq[kk] = *(const v2f*)(q + (size_t)(i0 + lm) * DTOT + ho + kk * 4 + lk);

  for (int jt = wave; jt < NTOK / 16; jt += 8) {
    const int j0 = jt * 16;
    v8f acc = {};
#pragma unroll
    for (int kk = 0; kk < 12; ++kk) {
      // B[kd][n] = k[j0+n][ho+kd]: contiguous v2f along kd within a lane
      v2f bf = *(const v2f*)(kmat + (size_t)(j0 + lm) * DTOT + ho + kk * 4 + lk);
      acc = wmma4(aq[kk], bf, acc);
    }
#pragma unroll
    for (int r = 0; r < 8; ++r) P[r + (hi << 3)][j0 + lm] = acc[r];
  }
  __syncthreads();

  // ---- softmax over j (16 threads per row), pair bias fused
  {
    const int r = tid >> 4;
    const int c = tid & 15;
    float lmax = -3.4e38f;
    for (int j = c; j < NTOK; j += 16) {
      float x = P[r][j] + pb[((size_t)(i0 + r) * NTOK + j) * NHEAD + h];
      P[r][j] = x;
      lmax = fmaxf(lmax, x);
    }
    red[tid] = lmax;
    __syncthreads();
    if (c == 0) {
      float m = red[tid];
      for (int t = 1; t < 16; ++t) m = fmaxf(m, red[tid + t]);
      rowmax[r] = m;
    }
    __syncthreads();
    const float m = rowmax[r];
    float lsum = 0.f;
    for (int j = c; j < NTOK; j += 16) {
      float e = expf(P[r][j] - m);
      P[r][j] = e;
      lsum += e;
    }
    red[tid] = lsum;
    __syncthreads();
    if (c == 0) {
      float sm = 0.f;
      for (int t = 0; t < 16; ++t) sm += red[tid + t];
      rowsum[r] = sm;
    }
    __syncthreads();
  }

  // ---- P(16x768) @ V(768x48): K split over 8 waves, LDS atomic reduce
  {
    v8f acc[3] = {};
    const int kbase = wave * (NTOK / 8);  // 96 keys per wave
    for (int k = kbase; k < kbase + NTOK / 8; k += 4) {
      v2f af = *(const v2f*)&P[lm][k + lk];
#pragma unroll
      for (int t = 0; t < 3; ++t) {
        const float* vp = vmat + (size_t)(k + lk) * DTOT + ho + t * 16 + lm;
        v2f bf;
        bf[0] = vp[0];
        bf[1] = vp[DTOT];
        acc[t] = wmma4(af, bf, acc[t]);
      }
    }
#pragma unroll
    for (int t = 0; t < 3; ++t)
#pragma unroll
      for (int r = 0; r < 8; ++r)
        atomicAdd(&accbuf[r + (hi << 3)][t * 16 + lm], acc[t][r]);
  }
  __syncthreads();

  // ---- epilogue: o = g * (att @ v) / rowsum
  for (int idx = tid; idx < 16 * DHEAD; idx += 256) {
    const int r = idx / DHEAD;
    const int d = idx % DHEAD;
    const size_t oi = (size_t)(i0 + r) * DTOT + ho + d;
    o[oi] = accbuf[r][d] / rowsum[r] * g[oi];
  }
}

// ---------------------------------------------------------------------------
// Host orchestration. Workspace layout (floats), total ~60.2 MB:
// ---------------------------------------------------------------------------
extern "C" void kernel_launch(void* const* d_in, const int* in_sizes, int n_in,
                              void* d_out, int out_size, void* d_ws,
                              size_t ws_size, hipStream_t stream) {
  (void)in_sizes; (void)n_in; (void)out_size; (void)ws_size;
  const float* a         = (const float*)d_in[0];
  const float* s         = (const float*)d_in[1];
  const float* z         = (const float*)d_in[2];
  const float* beta      = (const float*)d_in[3];
  const float* ln_s_w    = (const float*)d_in[4];
  const float* ln_s_b    = (const float*)d_in[5];
  const float* ws_gate_w = (const float*)d_in[6];
  const float* ws_gate_b = (const float*)d_in[7];
  const float* bs_w      = (const float*)d_in[8];
  const float* wq_w      = (const float*)d_in[9];
  const float* wq_b      = (const float*)d_in[10];
  const float* wk_w      = (const float*)d_in[11];
  const float* wv_w      = (const float*)d_in[12];
  const float* wg_w      = (const float*)d_in[13];
  const float* wb_w      = (const float*)d_in[14];
  const float* wb_b      = (const float*)d_in[15];
  const float* wo_w      = (const float*)d_in[16];
  const float* ws_out_w  = (const float*)d_in[17];
  const float* ws_out_b  = (const float*)d_in[18];
  float* out = (float*)d_out;

  const size_t ND = (size_t)NTOK * DTOT;        // 589824
  float* ws = (float*)d_ws;
  float* s_n      = ws;                         // N*DS = 294912
  float* a_n      = s_n + (size_t)NTOK * DSNG;
  float* bias_lin = a_n + ND;
  float* a2       = bias_lin + ND;
  float* qb       = a2 + ND;
  float* kb       = qb + ND;
  float* vb       = kb + ND;
  float* gb       = vb + ND;
  float* og       = gb + ND;
  float* ob       = og + ND;
  float* pbuf     = ob + ND;                    // N*N*H = 9437184

  const dim3 blk(256);
  const dim3 gg(NTOK / 128, DTOT / 64);  // (6, 12)
  const float qscale = rsqrtf((float)DHEAD);

  // LayerNorms
  ln_kernel<<<NTOK, blk, 0, stream>>>(a, nullptr, nullptr, a_n, DTOT);
  ln_kernel<<<NTOK, blk, 0, stream>>>(s, ln_s_w, ln_s_b, s_n, DSNG);

  // adaLN: bias_lin = s_n@bs_w ; a2 = sigmoid(s_n@Wg + bg)*a_n + bias_lin
  gemm_kernel<<<gg, blk, 0, stream>>>(s_n, bs_w, nullptr, bias_lin, DSNG, DTOT,
                                      1.f, 0, nullptr, nullptr);
  gemm_kernel<<<gg, blk, 0, stream>>>(s_n, ws_gate_w, ws_gate_b, a2, DSNG, DTOT,
                                      1.f, 3, a_n, bias_lin);

  // Projections
  gemm_kernel<<<gg, blk, 0, stream>>>(a2, wq_w, wq_b, qb, DTOT, DTOT, qscale, 0,
                                      nullptr, nullptr);
  gemm_kernel<<<gg, blk, 0, stream>>>(a2, wk_w, nullptr, kb, DTOT, DTOT, 1.f, 0,
                                      nullptr, nullptr);
  gemm_kernel<<<gg, blk, 0, stream>>>(a2, wv_w, nullptr, vb, DTOT, DTOT, 1.f, 0,
                                      nullptr, nullptr);
  gemm_kernel<<<gg, blk, 0, stream>>>(a2, wg_w, nullptr, gb, DTOT, DTOT, 1.f, 1,
                                      nullptr, nullptr);

  // Output gate from raw s
  gemm_kernel<<<gg, blk, 0, stream>>>(s, ws_out_w, ws_out_b, og, DSNG, DTOT,
                                      1.f, 0, nullptr, nullptr);

  // Pair bias (streams z once)
  pair_bias_kernel<<<(NTOK * NTOK) / 128, blk, 0, stream>>>(z, wb_w, wb_b, beta,
                                                            pbuf);

  // Attention
  attn_kernel<<<dim3(NTOK / 16, NHEAD), blk, 0, stream>>>(qb, kb, vb, gb, pbuf,
                                                          ob);

  // Output projection * sigmoid(out gate)
  gemm_kernel<<<gg, blk, 0, stream>>>(ob, wo_w, nullptr, out, DTOT, DTOT, 1.f,
                                      2, og, nullptr);
}